// GNNModel_17549236371687
// MI455X (gfx1250) — compile-verified
//
#include <hip/hip_runtime.h>

typedef float v2f __attribute__((ext_vector_type(2)));
typedef float v8f __attribute__((ext_vector_type(8)));

// ---------------- init / degree ----------------

__global__ void k_zero(float* __restrict__ p, int n) {
  int i = blockIdx.x * blockDim.x + threadIdx.x;
  if (i < n) p[i] = 0.0f;
}

__global__ void k_fill1(float* __restrict__ p, int n) {
  int i = blockIdx.x * blockDim.x + threadIdx.x;
  if (i < n) p[i] = 1.0f;
}

// deg[dst] += 1 for every edge (deg pre-initialized to 1.0 for the self loop)
__global__ void k_degree(const int* __restrict__ ei, float* __restrict__ deg, int E) {
  int e = blockIdx.x * blockDim.x + threadIdx.x;
  if (e < E) atomicAdd(&deg[ei[E + e]], 1.0f);
}

// deg -> dinv = rsqrt(deg); deg >= 1 always (self loops), matches reference mask
__global__ void k_dinv(float* __restrict__ deg, int n) {
  int i = blockIdx.x * blockDim.x + threadIdx.x;
  if (i < n) deg[i] = rsqrtf(deg[i]);
}

// ---------------- WMMA node transforms ----------------
// t1[N,16] = x[N,2] @ W1[2,16]  via V_WMMA_F32_16X16X4_F32, K padded 2->4.
// A (16x4):  lanes 0-15 = M rows, vgpr pair = K {2*half, 2*half+1}
// B (4x16):  lanes 0-15 = N cols, vgpr pair = K {2*half, 2*half+1} (half = lane>>4)
// D (16x16): c[j] is row M = j + 8*half, col N = lane&15
__global__ void k_xform1_wmma(const float* __restrict__ x, const float* __restrict__ W1,
                              float* __restrict__ t1, int n) {
  const int lane = threadIdx.x & 31;
  const int wave = threadIdx.x >> 5;
  const int half = lane >> 4;
  const int m    = lane & 15;
  const int base = (blockIdx.x * (int)(blockDim.x >> 5) + wave) * 16;
  int node = base + m;
  if (node >= n) node = n - 1;        // keep EXEC all-ones; tail stores guarded below

  // branchless: K=0,1 lanes (half==0) carry real data, K=2,3 lanes are zeroed
  const float msk = (half == 0) ? 1.0f : 0.0f;
  v2f a, b;
  a.x = x[node * 2 + 0] * msk;
  a.y = x[node * 2 + 1] * msk;
  b.x = W1[m]      * msk;
  b.y = W1[16 + m] * msk;

  v8f c = {};
  c = __builtin_amdgcn_wmma_f32_16x16x4_f32(false, a, false, b, (short)0, c, false, false);

  if (base + 16 <= n) {               // wave-uniform fast path: whole tile in range
#pragma unroll
    for (int j = 0; j < 8; ++j)
      t1[(base + j + 8 * half) * 16 + m] = c[j];
  } else {                            // cold tail: per-row guards
#pragma unroll
    for (int j = 0; j < 8; ++j) {
      int row = base + j + 8 * half;
      if (row < n) t1[row * 16 + m] = c[j];
    }
  }
}

// t2[N,32] = h1[N,16] @ W2[16,32]: 4 chained K=4 WMMAs, two 16-wide N tiles.
__global__ void k_xform2_wmma(const float* __restrict__ h1, const float* __restrict__ W2,
                              float* __restrict__ t2, int n) {
  const int lane = threadIdx.x & 31;
  const int wave = threadIdx.x >> 5;
  const int half = lane >> 4;
  const int m    = lane & 15;
  const int base = (blockIdx.x * (int)(blockDim.x >> 5) + wave) * 16;
  int node = base + m;
  if (node >= n) node = n - 1;

  v8f c0 = {}, c1 = {};
#pragma unroll
  for (int s = 0; s < 4; ++s) {
    const int k = 4 * s + 2 * half;
    v2f a, b0, b1;
    a.x  = h1[node * 16 + k + 0];
    a.y  = h1[node * 16 + k + 1];
    b0.x = W2[(k + 0) * 32 + m];
    b0.y = W2[(k + 1) * 32 + m];
    b1.x = W2[(k + 0) * 32 + 16 + m];
    b1.y = W2[(k + 1) * 32 + 16 + m];
    c0 = __builtin_amdgcn_wmma_f32_16x16x4_f32(false, a, false, b0, (short)0, c0, false, false);
    c1 = __builtin_amdgcn_wmma_f32_16x16x4_f32(false, a, false, b1, (short)0, c1, false, false);
  }

  if (base + 16 <= n) {               // wave-uniform fast path
#pragma unroll
    for (int j = 0; j < 8; ++j) {
      int row = base + j + 8 * half;
      t2[row * 32 + m]      = c0[j];
      t2[row * 32 + 16 + m] = c1[j];
    }
  } else {
#pragma unroll
    for (int j = 0; j < 8; ++j) {
      int row = base + j + 8 * half;
      if (row < n) {
        t2[row * 32 + m]      = c0[j];
        t2[row * 32 + 16 + m] = c1[j];
      }
    }
  }
}

// ---------------- edge scatter (the bandwidth-bound core) ----------------
// One lane per (edge, feature): F-lane groups give coalesced gathers and
// coalesced atomic segments on the destination row.
template <int F>
__global__ void k_scatter(const int* __restrict__ ei, const float* __restrict__ dinv,
                          const float* __restrict__ t, float* __restrict__ agg, int E) {
  int tid = blockIdx.x * blockDim.x + threadIdx.x;
  int e = tid / F;
  int f = tid % F;
  if (e < E) {
    int s = ei[e];
    int d = ei[E + e];
    float nrm = dinv[s] * dinv[d];
    atomicAdd(&agg[d * F + f], t[s * F + f] * nrm);
  }
}

// agg = relu(agg + t*dinv^2 + b)   (self-loop term + bias + activation, fused)
template <int F>
__global__ void k_finish(const float* __restrict__ t, const float* __restrict__ dinv,
                         const float* __restrict__ b, float* __restrict__ agg, int n) {
  int tid = blockIdx.x * blockDim.x + threadIdx.x;
  int node = tid / F;
  int f    = tid % F;
  if (node < n) {
    float di = dinv[node];
    float v = agg[tid] + t[tid] * di * di + b[f];
    agg[tid] = fmaxf(v, 0.0f);
  }
}

// ---------------- pooling + head ----------------

__global__ void k_pool(const float* __restrict__ h2, const int* __restrict__ batch,
                       float* __restrict__ gsum, float* __restrict__ gcnt, int n) {
  int tid = blockIdx.x * blockDim.x + threadIdx.x;
  int node = tid >> 5;
  int f    = tid & 31;
  if (node < n) {
    int g = batch[node];
    atomicAdd(&gsum[g * 32 + f], h2[tid]);
    if (f == 0) atomicAdd(&gcnt[g], 1.0f);
  }
}

__global__ void k_head(const float* __restrict__ gsum, const float* __restrict__ gcnt,
                       const float* __restrict__ Wfc1, const float* __restrict__ bfc1,
                       const float* __restrict__ Wfc2, const float* __restrict__ bfc2,
                       float* __restrict__ out, int G) {
  int g = blockIdx.x * blockDim.x + threadIdx.x;
  if (g < G) {
    float inv = 1.0f / fmaxf(gcnt[g], 1.0f);
    float p[32];
#pragma unroll
    for (int f = 0; f < 32; ++f) p[f] = gsum[g * 32 + f] * inv;
    float acc = bfc2[0];
#pragma unroll
    for (int j = 0; j < 16; ++j) {
      float h = bfc1[j];
#pragma unroll
      for (int f = 0; f < 32; ++f) h += p[f] * Wfc1[f * 16 + j];
      acc += fmaxf(h, 0.0f) * Wfc2[j];
    }
    out[g] = acc;
  }
}

// ---------------- launch ----------------

extern "C" void kernel_launch(void* const* d_in, const int* in_sizes, int n_in,
                              void* d_out, int out_size, void* d_ws, size_t ws_size,
                              hipStream_t stream) {
  (void)n_in; (void)ws_size;
  const float* x     = (const float*)d_in[0];
  const int*   ei    = (const int*)d_in[1];   // [2,E]: row0=src, row1=dst
  const int*   batch = (const int*)d_in[2];
  const float* W1    = (const float*)d_in[3];
  const float* b1    = (const float*)d_in[4];
  const float* W2    = (const float*)d_in[5];
  const float* b2    = (const float*)d_in[6];
  const float* Wfc1  = (const float*)d_in[7];
  const float* bfc1  = (const float*)d_in[8];
  const float* Wfc2  = (const float*)d_in[9];
  const float* bfc2  = (const float*)d_in[10];
  float* out = (float*)d_out;

  const int N = in_sizes[2];       // 200000 nodes
  const int E = in_sizes[1] / 2;   // 6.4M edges
  const int G = out_size;          // 1000 graphs

  // workspace layout (floats): ~97N + 33G = ~78 MB
  float* ws   = (float*)d_ws;
  float* deg  = ws;                        // N       (becomes dinv in place)
  float* t1   = deg  + (size_t)N;          // 16N
  float* agg1 = t1   + (size_t)N * 16;     // 16N     (becomes h1 in place)
  float* t2   = agg1 + (size_t)N * 16;     // 32N
  float* agg2 = t2   + (size_t)N * 32;     // 32N     (becomes h2 in place)
  float* gsum = agg2 + (size_t)N * 32;     // 32G
  float* gcnt = gsum + (size_t)G * 32;     // G       (contiguous with gsum)

  const int B = 256;
  auto cdiv = [](long long a, int b) { return (int)((a + b - 1) / b); };

  // re-initialize every call (graph replay must be deterministic)
  k_fill1<<<cdiv(N, B), B, 0, stream>>>(deg, N);
  k_zero <<<cdiv((long long)N * 16, B), B, 0, stream>>>(agg1, N * 16);
  k_zero <<<cdiv((long long)N * 32, B), B, 0, stream>>>(agg2, N * 32);
  k_zero <<<cdiv((long long)G * 33, B), B, 0, stream>>>(gsum, G * 33);

  k_degree<<<cdiv(E, B), B, 0, stream>>>(ei, deg, E);
  k_dinv  <<<cdiv(N, B), B, 0, stream>>>(deg, N);

  const int nodesPerBlk = (B / 32) * 16;   // 8 waves x 16 rows
  const int gW = cdiv(N, nodesPerBlk);

  // layer 1
  k_xform1_wmma<<<gW, B, 0, stream>>>(x, W1, t1, N);
  k_scatter<16><<<cdiv((long long)E * 16, B), B, 0, stream>>>(ei, deg, t1, agg1, E);
  k_finish <16><<<cdiv((long long)N * 16, B), B, 0, stream>>>(t1, deg, b1, agg1, N);

  // layer 2
  k_xform2_wmma<<<gW, B, 0, stream>>>(agg1, W2, t2, N);
  k_scatter<32><<<cdiv((long long)E * 32, B), B, 0, stream>>>(ei, deg, t2, agg2, E);
  k_finish <32><<<cdiv((long long)N * 32, B), B, 0, stream>>>(t2, deg, b2, agg2, N);

  // mean pool + MLP head
  k_pool<<<cdiv((long long)N * 32, B), B, 0, stream>>>(agg2, batch, gsum, gcnt, N);
  k_head<<<cdiv(G, B), B, 0, stream>>>(gsum, gcnt, Wfc1, bfc1, Wfc2, bfc2, out, G);
}